// GCL_global_28681791603392
// MI455X (gfx1250) — compile-verified
//
#include <hip/hip_runtime.h>

typedef __attribute__((ext_vector_type(2))) float v2f;
typedef __attribute__((ext_vector_type(8))) float v8f;

#define N_NODES 100000
#define FEATS   128
#define N_EDGES 1600000
#define WPB     8                      // waves per block (256 threads)
#define NTILES  (N_NODES / 16)         // 6250 row-tiles of 16

// ---------------------------------------------------------------- zero d_out
__global__ __launch_bounds__(256) void gcl_zero_kernel(float4* __restrict__ out, int n4) {
  int i = blockIdx.x * 256 + threadIdx.x;
  if (i < n4) out[i] = make_float4(0.f, 0.f, 0.f, 0.f);
}

// ------------------------------------------------- hp = (h @ wh) * norm  (WMMA f32)
__global__ __launch_bounds__(256) void gcl_project_h_kernel(
    const float* __restrict__ h, const float* __restrict__ wh,
    const float* __restrict__ norm, float* __restrict__ hp) {
  __shared__ float ldsw[FEATS * FEATS];          // 64 KB weight tile
  {
    float4* d = (float4*)ldsw;
    const float4* s = (const float4*)wh;
    for (int i = threadIdx.x; i < FEATS * FEATS / 4; i += 256) d[i] = s[i];
  }
  __syncthreads();

  int wave = threadIdx.x >> 5;
  int tile = __builtin_amdgcn_readfirstlane(blockIdx.x * WPB + wave);
  if (tile >= NTILES) return;                    // scalar branch, EXEC stays full

  int lane = threadIdx.x & 31;
  int hi   = lane >> 4;
  int lr   = lane & 15;
  int rowbase = tile * 16;

  // A-tile fragments: lane half 'hi' holds K = 4*kk + 2*hi + {0,1}  (ISA 16x4 f32 layout)
  v2f afrag[32];
  const float* hrow = h + (size_t)(rowbase + lr) * FEATS;
  #pragma unroll
  for (int kk = 0; kk < 32; ++kk) {
    float2 t = ((const float2*)(hrow + kk * 4))[hi];
    afrag[kk].x = t.x; afrag[kk].y = t.y;
  }
  float nrm[8];
  #pragma unroll
  for (int r = 0; r < 8; ++r) nrm[r] = norm[rowbase + r + 8 * hi];

  #pragma unroll
  for (int nt = 0; nt < 8; ++nt) {
    int ncol = nt * 16 + lr;
    v8f acc = {};
    #pragma unroll
    for (int kk = 0; kk < 32; ++kk) {
      int k0 = kk * 4 + 2 * hi;                  // B mirrors A's K striping
      v2f b;
      b.x = ldsw[(k0 + 0) * FEATS + ncol];
      b.y = ldsw[(k0 + 1) * FEATS + ncol];
      acc = __builtin_amdgcn_wmma_f32_16x16x4_f32(
          false, afrag[kk], false, b, (short)0, acc, false, false);
    }
    #pragma unroll
    for (int r = 0; r < 8; ++r) {                // C layout: row = r + 8*hi, col = lr
      int row = rowbase + r + 8 * hi;
      hp[(size_t)row * FEATS + ncol] = acc[r] * nrm[r];
    }
  }
}

// ------------------------------------------- segment_sum: out[dst] += hp[src]  (L2 atomics)
__global__ __launch_bounds__(256) void gcl_edge_scatter_kernel(
    const float* __restrict__ hp, const int* __restrict__ src,
    const int* __restrict__ dst, float* __restrict__ out) {
  int e = (blockIdx.x * 256 + threadIdx.x) >> 5;  // one wave per edge
  if (e >= N_EDGES) return;
  int lane = threadIdx.x & 31;                    // lane -> float4 feature chunk
  int s = src[e];
  int d = dst[e];
  float4 v = ((const float4*)(hp + (size_t)s * FEATS))[lane];
  float* op = out + (size_t)d * FEATS + lane * 4;
  unsafeAtomicAdd(op + 0, v.x);
  unsafeAtomicAdd(op + 1, v.y);
  unsafeAtomicAdd(op + 2, v.z);
  unsafeAtomicAdd(op + 3, v.w);
}

// ---------------- out = relu(agg*norm + bh + (m@wm) + bm)  (WMMA f32, fused epilogue)
__global__ __launch_bounds__(256) void gcl_project_m_combine_kernel(
    const float* __restrict__ m, const float* __restrict__ wm,
    const float* __restrict__ norm, const float* __restrict__ bh,
    const float* __restrict__ bm, float* __restrict__ out) {
  __shared__ float ldsw[FEATS * FEATS];
  {
    float4* d = (float4*)ldsw;
    const float4* s = (const float4*)wm;
    for (int i = threadIdx.x; i < FEATS * FEATS / 4; i += 256) d[i] = s[i];
  }
  __syncthreads();

  int wave = threadIdx.x >> 5;
  int tile = __builtin_amdgcn_readfirstlane(blockIdx.x * WPB + wave);
  if (tile >= NTILES) return;

  int lane = threadIdx.x & 31;
  int hi   = lane >> 4;
  int lr   = lane & 15;
  int rowbase = tile * 16;

  v2f afrag[32];
  const float* mrow = m + (size_t)(rowbase + lr) * FEATS;
  #pragma unroll
  for (int kk = 0; kk < 32; ++kk) {
    float2 t = ((const float2*)(mrow + kk * 4))[hi];
    afrag[kk].x = t.x; afrag[kk].y = t.y;
  }
  float nrm[8];
  #pragma unroll
  for (int r = 0; r < 8; ++r) nrm[r] = norm[rowbase + r + 8 * hi];

  #pragma unroll
  for (int nt = 0; nt < 8; ++nt) {
    int ncol = nt * 16 + lr;
    float bias = bh[ncol] + bm[ncol];
    v8f acc = {};
    #pragma unroll
    for (int kk = 0; kk < 32; ++kk) {
      int k0 = kk * 4 + 2 * hi;
      v2f b;
      b.x = ldsw[(k0 + 0) * FEATS + ncol];
      b.y = ldsw[(k0 + 1) * FEATS + ncol];
      acc = __builtin_amdgcn_wmma_f32_16x16x4_f32(
          false, afrag[kk], false, b, (short)0, acc, false, false);
    }
    #pragma unroll
    for (int r = 0; r < 8; ++r) {
      int row = rowbase + r + 8 * hi;
      size_t idx = (size_t)row * FEATS + ncol;
      float v = out[idx] * nrm[r] + acc[r] + bias;   // agg*norm + m@wm + (bh+bm)
      out[idx] = fmaxf(v, 0.f);                      // relu
    }
  }
}

// --------------------------------------------------------------------- launch
extern "C" void kernel_launch(void* const* d_in, const int* in_sizes, int n_in,
                              void* d_out, int out_size, void* d_ws, size_t ws_size,
                              hipStream_t stream) {
  const float* h    = (const float*)d_in[0];
  const float* m    = (const float*)d_in[1];
  const float* norm = (const float*)d_in[2];
  const int*   src  = (const int*)d_in[3];
  const int*   dst  = (const int*)d_in[4];
  const float* wh   = (const float*)d_in[5];
  const float* wm   = (const float*)d_in[6];
  const float* bh   = (const float*)d_in[7];
  const float* bm   = (const float*)d_in[8];
  float* out = (float*)d_out;
  float* hp  = (float*)d_ws;                       // 100000*128 f32 = 51.2 MB scratch

  int n4 = N_NODES * FEATS / 4;
  gcl_zero_kernel<<<(n4 + 255) / 256, 256, 0, stream>>>((float4*)out, n4);

  int gblocks = (NTILES + WPB - 1) / WPB;          // 782
  gcl_project_h_kernel<<<gblocks, 256, 0, stream>>>(h, wh, norm, hp);

  int eblocks = (N_EDGES * 32) / 256;              // 200000, one wave per edge
  gcl_edge_scatter_kernel<<<eblocks, 256, 0, stream>>>(hp, src, dst, out);

  gcl_project_m_combine_kernel<<<gblocks, 256, 0, stream>>>(m, wm, norm, bh, bm, out);
}